// TSM_6330781795061
// MI455X (gfx1250) — compile-verified
//
#include <hip/hip_runtime.h>

// Temporal-shift "copy with zeroed planes":
//   in/out shape (B=1440, C=64, H=36, W=36) fp32, clips of T=180, fold=21.
//   out == in, except plane (b,c) is zero when
//     (b%180 ==   0 && c < 21)  ||
//     (b%180 == 179 && 21 <= c < 42)
//
// Pure streaming: 478 MB in + 478 MB out -> ~41 us at 23.3 TB/s.
// Strategy: one float4 per thread, b128 NT loads/stores, predicate fused.

typedef __attribute__((ext_vector_type(4))) float f4;

#define HW4   324u   // 36*36/4 float4 per (b,c) plane
#define CMASK 63u    // C = 64
#define CSH   6
#define TLEN  180u
#define FOLD  21u

__global__ __launch_bounds__(256) void tsm_shift_copy_kernel(
    const f4* __restrict__ in, f4* __restrict__ out, unsigned n4)
{
    unsigned idx = blockIdx.x * 256u + threadIdx.x;
    if (idx >= n4) return;

    // Recover (b, c) of the 1296-float plane this float4 belongs to.
    unsigned plane = idx / HW4;          // magic-number mul, no real div
    unsigned c     = plane & CMASK;
    unsigned b     = plane >> CSH;
    unsigned t     = b % TLEN;           // time index within clip

    // Stream the data through with last-touch (non-temporal) hints:
    // dataset (478 MB) >> L2 (192 MB), each byte touched exactly once.
    f4 v = __builtin_nontemporal_load(in + idx);

    bool zero = ((t == 0u)        & (c < FOLD)) |
                ((t == TLEN - 1u) & (c >= FOLD) & (c < 2u * FOLD));
    if (zero) v = (f4){0.f, 0.f, 0.f, 0.f};   // lowers to v_cndmask, no branch

    __builtin_nontemporal_store(v, out + idx);
}

extern "C" void kernel_launch(void* const* d_in, const int* in_sizes, int n_in,
                              void* d_out, int out_size, void* d_ws, size_t ws_size,
                              hipStream_t stream)
{
    const f4* in  = (const f4*)d_in[0];
    f4*       out = (f4*)d_out;

    // 1440*64*36*36 = 119,439,360 floats -> 29,859,840 float4 (HW=1296 is /4)
    unsigned n4     = (unsigned)(in_sizes[0] >> 2);
    unsigned blocks = (n4 + 255u) / 256u;

    tsm_shift_copy_kernel<<<blocks, 256, 0, stream>>>(in, out, n4);
}